// GCN_89378269430352
// MI455X (gfx1250) — compile-verified
//
#include <hip/hip_runtime.h>
#include <hip/hip_bf16.h>

typedef __attribute__((ext_vector_type(2))) float v2f;
typedef __attribute__((ext_vector_type(8))) float v8f;

#define GAT_N    50000
#define GAT_E    800000
#define GAT_DIN  128
#define GAT_DHID 64
#define GAT_DOUT 32
#define NEG_SLOPE 0.2f

// ---------------- monotonic float <-> uint encoding for atomic max ----------
__device__ __forceinline__ unsigned flipf(float f) {
  unsigned u = __float_as_uint(f);
  return (u & 0x80000000u) ? ~u : (u | 0x80000000u);
}
__device__ __forceinline__ float unflipf(unsigned u) {
  unsigned b = (u & 0x80000000u) ? (u & 0x7FFFFFFFu) : ~u;
  return __uint_as_float(b);
}

// ---------------- dense projection: C[M,NCOL] = A[M,K] * B[K,NCOL] ----------
// One wave computes a 16-row stripe across all NCOL columns using
// V_WMMA_F32_16X16X4_F32 (full fp32 matrix core path).
// A 16x4 layout (ISA 7.12.2): lanes 0-15 hold K=0,1 in v0,v1; lanes 16-31 K=2,3.
// B 4x16 layout: symmetric (rows striped across lanes within a VGPR).
// C/D 16x16 f32: VGPR r -> M = r + 8*(lane>=16), N = lane%16.
template<int K, int NCOL>
__global__ __launch_bounds__(256)
void wmma_gemm_kernel(const float* __restrict__ A, const float* __restrict__ B,
                      float* __restrict__ C, int Mrows) {
  constexpr int NT = NCOL / 16;
  int wave = (int)((blockIdx.x * blockDim.x + threadIdx.x) >> 5);
  int lane = (int)(threadIdx.x & 31);
  int m0 = wave * 16;
  if (m0 >= Mrows) return;                 // wave-uniform: EXEC stays all-ones
  int half = lane >> 4;                    // 0: K pair (k,k+1), 1: (k+2,k+3)
  int lr = lane & 15;

  v8f acc[NT] = {};
  const float* arow = A + (size_t)(m0 + lr) * K;

  for (int k = 0; k < K; k += 4) {
    int ka = k + 2 * half;
    v2f a;
    a.x = arow[ka];
    a.y = arow[ka + 1];
    const float* b0 = B + (size_t)ka * NCOL;
    const float* b1 = B + (size_t)(ka + 1) * NCOL;
#pragma unroll
    for (int t = 0; t < NT; ++t) {
      v2f b;
      b.x = b0[t * 16 + lr];
      b.y = b1[t * 16 + lr];
      acc[t] = __builtin_amdgcn_wmma_f32_16x16x4_f32(
          /*neg_a=*/false, a, /*neg_b=*/false, b,
          /*c_mod=*/(short)0, acc[t], /*reuse_a=*/false, /*reuse_b=*/false);
    }
  }

  int row = m0 + half * 8;
#pragma unroll
  for (int t = 0; t < NT; ++t) {
#pragma unroll
    for (int r = 0; r < 8; ++r) {
      C[(size_t)(row + r) * NCOL + t * 16 + lr] = acc[t][r];
    }
  }
}

// ---------------- per-node attention logits: el/er = <feat[n,h,:], a[h,:]> ---
template<int F>
__global__ __launch_bounds__(256)
void node_attn_kernel(const float* __restrict__ feat,
                      const float* __restrict__ al, const float* __restrict__ ar,
                      float* __restrict__ el, float* __restrict__ er, int n_nodes) {
  int wave = (int)((blockIdx.x * blockDim.x + threadIdx.x) >> 5);
  int lane = (int)(threadIdx.x & 31);
  if (wave >= n_nodes) return;
  const float* frow = feat + (size_t)wave * (2 * F);
#pragma unroll
  for (int h = 0; h < 2; ++h) {
    float sl = 0.f, sr = 0.f;
    for (int f = lane; f < F; f += 32) {
      float v = frow[h * F + f];
      sl += v * al[h * F + f];
      sr += v * ar[h * F + f];
    }
#pragma unroll
    for (int off = 16; off > 0; off >>= 1) {
      sl += __shfl_xor(sl, off, 32);
      sr += __shfl_xor(sr, off, 32);
    }
    if (lane == 0) {
      el[wave * 2 + h] = sl;
      er[wave * 2 + h] = sr;
    }
  }
}

// ---------------- edge pass A: e = leaky_relu(el[src]+er[dst]); segment max --
__global__ __launch_bounds__(256)
void edge_logits_kernel(const int* __restrict__ src, const int* __restrict__ dst,
                        const float* __restrict__ el, const float* __restrict__ er,
                        float* __restrict__ e, unsigned* __restrict__ menc, int nE) {
  int i = (int)(blockIdx.x * blockDim.x + threadIdx.x);
  if (i >= nE) return;
  int s = src[i], d = dst[i];
#pragma unroll
  for (int h = 0; h < 2; ++h) {
    float v = el[s * 2 + h] + er[d * 2 + h];
    v = (v > 0.f) ? v : NEG_SLOPE * v;
    e[(size_t)i * 2 + h] = v;
    atomicMax(&menc[d * 2 + h], flipf(v));
  }
}

// ---------------- edge pass B: alpha = exp(e - m[dst]); segment sum ----------
__global__ __launch_bounds__(256)
void edge_alpha_kernel(const int* __restrict__ dst, const unsigned* __restrict__ menc,
                       float* __restrict__ e_alpha, float* __restrict__ denom, int nE) {
  int i = (int)(blockIdx.x * blockDim.x + threadIdx.x);
  if (i >= nE) return;
  int d = dst[i];
#pragma unroll
  for (int h = 0; h < 2; ++h) {
    float m = unflipf(menc[d * 2 + h]);
    float a = __expf(e_alpha[(size_t)i * 2 + h] - m);
    e_alpha[(size_t)i * 2 + h] = a;
    atomicAdd(&denom[d * 2 + h], a);
  }
}

// ---------------- edge pass C: agg[dst] += feat[src] * alpha/denom -----------
// One wave per edge; 32 lanes cover 2*F feature columns (VEC each).
template<int F>
__global__ __launch_bounds__(256)
void edge_agg_kernel(const int* __restrict__ src, const int* __restrict__ dst,
                     const float* __restrict__ feat, const float* __restrict__ alpha,
                     const float* __restrict__ denom, float* __restrict__ agg, int nE) {
  constexpr int C = 2 * F;
  constexpr int VEC = C / 32;
  int wave = (int)((blockIdx.x * blockDim.x + threadIdx.x) >> 5);
  int lane = (int)(threadIdx.x & 31);
  if (wave >= nE) return;
  int s = src[wave], d = dst[wave];
  int col = lane * VEC;
  int h = (col >= F) ? 1 : 0;
  float sc = alpha[(size_t)wave * 2 + h] / denom[d * 2 + h];
  const float* fsrc = feat + (size_t)s * C + col;
  float* ad = agg + (size_t)d * C + col;
#pragma unroll
  for (int j = 0; j < VEC; ++j) {
    atomicAdd(&ad[j], fsrc[j] * sc);
  }
}

// ---------------- head-mean + bias + relu ------------------------------------
template<int F>
__global__ __launch_bounds__(256)
void finish_kernel(const float* __restrict__ agg, const float* __restrict__ bias,
                   float* __restrict__ out, int n_nodes) {
  int i = (int)(blockIdx.x * blockDim.x + threadIdx.x);
  if (i >= n_nodes * F) return;
  int n = i / F, f = i % F;
  float v = 0.5f * (agg[(size_t)n * 2 * F + f] + bias[f] +
                    agg[(size_t)n * 2 * F + F + f] + bias[F + f]);
  out[i] = (v > 0.f) ? v : 0.f;
}

static inline int cdiv(long long a, long long b) { return (int)((a + b - 1) / b); }

extern "C" void kernel_launch(void* const* d_in, const int* in_sizes, int n_in,
                              void* d_out, int out_size, void* d_ws, size_t ws_size,
                              hipStream_t stream) {
  (void)in_sizes; (void)n_in; (void)out_size; (void)ws_size;
  const float* x   = (const float*)d_in[0];
  const int*  esrc = (const int*)d_in[1];
  const int*  edst = (const int*)d_in[2];
  const float* W1  = (const float*)d_in[3];
  const float* al1 = (const float*)d_in[4];
  const float* ar1 = (const float*)d_in[5];
  const float* b1  = (const float*)d_in[6];
  const float* W2  = (const float*)d_in[7];
  const float* al2 = (const float*)d_in[8];
  const float* ar2 = (const float*)d_in[9];
  const float* b2  = (const float*)d_in[10];

  const int N = GAT_N, E = GAT_E;

  // ---- workspace layout (floats) ----
  float* ws = (float*)d_ws;
  size_t o = 0;
  float*    feat1  = ws + o; o += (size_t)N * 128;
  float*    el1    = ws + o; o += (size_t)N * 2;
  float*    er1    = ws + o; o += (size_t)N * 2;
  unsigned* m1     = (unsigned*)(ws + o); o += (size_t)N * 2;
  float*    denom1 = ws + o; o += (size_t)N * 2;
  float*    e1     = ws + o; o += (size_t)E * 2;   // logits, overwritten by alpha
  float*    agg1   = ws + o; o += (size_t)N * 128;
  float*    h1     = ws + o; o += (size_t)N * 64;
  float*    feat2  = ws + o; o += (size_t)N * 64;
  float*    el2    = ws + o; o += (size_t)N * 2;
  float*    er2    = ws + o; o += (size_t)N * 2;
  unsigned* m2     = (unsigned*)(ws + o); o += (size_t)N * 2;
  float*    denom2 = ws + o; o += (size_t)N * 2;
  float*    e2     = ws + o; o += (size_t)E * 2;
  float*    agg2   = ws + o; o += (size_t)N * 64;

  // ---- zero the accumulators (capture-legal memset nodes) ----
  hipMemsetAsync(m1,     0, (size_t)N * 2 * 4, stream);
  hipMemsetAsync(denom1, 0, (size_t)N * 2 * 4, stream);
  hipMemsetAsync(agg1,   0, (size_t)N * 128 * 4, stream);
  hipMemsetAsync(m2,     0, (size_t)N * 2 * 4, stream);
  hipMemsetAsync(denom2, 0, (size_t)N * 2 * 4, stream);
  hipMemsetAsync(agg2,   0, (size_t)N * 64 * 4, stream);

  const int BLK = 256;
  // ================= Layer 1 =================
  {
    int waves = N / 16;                                  // 3125 row stripes
    wmma_gemm_kernel<128, 128><<<cdiv((long long)waves * 32, BLK), BLK, 0, stream>>>(
        x, W1, feat1, N);
    node_attn_kernel<64><<<cdiv((long long)N * 32, BLK), BLK, 0, stream>>>(
        feat1, al1, ar1, el1, er1, N);
    edge_logits_kernel<<<cdiv(E, BLK), BLK, 0, stream>>>(esrc, edst, el1, er1, e1, m1, E);
    edge_alpha_kernel<<<cdiv(E, BLK), BLK, 0, stream>>>(edst, m1, e1, denom1, E);
    edge_agg_kernel<64><<<cdiv((long long)E * 32, BLK), BLK, 0, stream>>>(
        esrc, edst, feat1, e1, denom1, agg1, E);
    finish_kernel<64><<<cdiv((long long)N * 64, BLK), BLK, 0, stream>>>(agg1, b1, h1, N);
  }
  // ================= Layer 2 =================
  {
    int waves = N / 16;
    wmma_gemm_kernel<64, 64><<<cdiv((long long)waves * 32, BLK), BLK, 0, stream>>>(
        h1, W2, feat2, N);
    node_attn_kernel<32><<<cdiv((long long)N * 32, BLK), BLK, 0, stream>>>(
        feat2, al2, ar2, el2, er2, N);
    edge_logits_kernel<<<cdiv(E, BLK), BLK, 0, stream>>>(esrc, edst, el2, er2, e2, m2, E);
    edge_alpha_kernel<<<cdiv(E, BLK), BLK, 0, stream>>>(edst, m2, e2, denom2, E);
    edge_agg_kernel<32><<<cdiv((long long)E * 32, BLK), BLK, 0, stream>>>(
        esrc, edst, feat2, e2, denom2, agg2, E);
    finish_kernel<32><<<cdiv((long long)N * 32, BLK), BLK, 0, stream>>>(
        agg2, b2, (float*)d_out, N);
  }
}